// LRU_76270029242927
// MI455X (gfx1250) — compile-verified
//
#include <hip/hip_runtime.h>

// ---------------------------------------------------------------------------
// LRU layer for MI455X (gfx1250, wave32).
//   Bu  = x @ (gamma*B)^T            (fp32 WMMA GEMM, complex output planes)
//   s_l = Lambda*s_{l-1} + Bu_l      (chunked parallel scan, 32 chunks x 128)
//   y   = s_re@C_re^T - s_im@C_im^T + x*D   (fused scan+WMMA GEMM)
// Precision: reference is fp32 -> V_WMMA_F32_16X16X4_F32.
// Data movement: Tensor Data Mover double-buffers operand panels into LDS in
// K-contiguous layout so every WMMA fragment is one aligned ds_load_b64.
// ---------------------------------------------------------------------------

#define BSZ    8
#define SEQL   4096
#define NDIM   256
#define HDIM   256
#define NCHUNK 32
#define CHUNK  128          // SEQL / NCHUNK
#define MTOT   (BSZ * SEQL) // 32768
#define PITCH  260          // A-strip LDS pitch in dwords (256 + 4 TDM pad)
#define KC     32           // K-panel depth per TDM transfer
#define KP     34           // B/C panel LDS pitch in dwords (32 + 2 TDM pad)

typedef __attribute__((ext_vector_type(2))) float v2f;
typedef __attribute__((ext_vector_type(8))) float v8f;
typedef unsigned int u32x4 __attribute__((ext_vector_type(4)));
typedef int i32x8 __attribute__((ext_vector_type(8)));
typedef int i32x4 __attribute__((ext_vector_type(4)));

__device__ __forceinline__ v8f wmma_f32(v2f a, v2f b, v8f c) {
  // 8 args: (neg_a, A, neg_b, B, c_mod, C, reuse_a, reuse_b)
  return __builtin_amdgcn_wmma_f32_16x16x4_f32(false, a, false, b, (short)0, c,
                                               false, false);
}

__device__ __forceinline__ v2f ldv2(const float* p) {  // 8B-aligned pair load
  return *(const v2f*)p;
}

// LDS byte offset of a __shared__ object (flat addr low 32 bits == LDS offset).
__device__ __forceinline__ unsigned lds_off(const void* p) {
  return (unsigned)(unsigned long long)p;
}

// ---------------------------------------------------------------------------
// TDM: async 2D tile load (rows x cols f32) global -> LDS with per-row LDS
// padding (pitch = cols + pad dwords). D# packing per CDNA5 ISA 8.3-8.5.
// pad_icode: pad every 2^(code+1) dwords; pad_acode: pad (code+1) dwords.
// 6-arg builtin on this toolchain: (g0, g1, g2, g3, g4, cpol).
// ---------------------------------------------------------------------------
__device__ __forceinline__ void tdm_load_2d(unsigned lds_byte, const float* gptr,
                                            int cols, int rows, int stride,
                                            int pad_icode, int pad_acode) {
  unsigned long long ga = (unsigned long long)gptr;
  u32x4 g0;
  g0[0] = 1u;                                   // count=1, user descriptor
  g0[1] = lds_byte;                             // lds_addr
  g0[2] = (unsigned)(ga & 0xFFFFFFFFu);         // global_addr[31:0]
  g0[3] = (unsigned)((ga >> 32) & 0x01FFFFFFu)  // global_addr[56:32]
        | (2u << 30);                           // type = 2 (image)
  i32x8 g1;
  g1[0] = (2 << 16)                             // data_size = 4 bytes
        | (1 << 20)                             // pad_enable
        | (pad_icode << 22)                     // pad interval
        | (pad_acode << 25);                    // pad amount
  g1[1] = (cols & 0xFFFF) << 16;                // tensor_dim0 lo16
  g1[2] = ((rows & 0xFFFF) << 16);              // dim0 hi=0 | tensor_dim1 lo16
  g1[3] = (cols << 16);                         // dim1 hi=0 | tile_dim0
  g1[4] = (rows & 0xFFFF);                      // tile_dim1 = rows, tile_dim2=0
  g1[5] = stride;                               // tensor_dim0_stride lo32
  g1[6] = 0;                                    // stride hi | dim1_stride lo
  g1[7] = 0;
  i32x4 z4 = {0, 0, 0, 0};
  i32x8 z8 = {0, 0, 0, 0, 0, 0, 0, 0};
  __builtin_amdgcn_tensor_load_to_lds(g0, g1, z4, z4, z8, 0);
}

// panel: KC-column slice of a 256x256 matrix, rows n-major -> LDS pitch KP
__device__ __forceinline__ void tdm_load_panel(unsigned lds_byte,
                                               const float* gptr) {
  tdm_load_2d(lds_byte, gptr, KC, NDIM, NDIM, /*icode=*/4, /*acode=*/1);
}

// ---------------------------------------------------------------------------
// K1: parameter prep (no transposes needed anymore — operand panels are
// consumed in native memory order). Scale B by gamma, negate C_im, emit
// Lambda.
// ---------------------------------------------------------------------------
__global__ __launch_bounds__(256) void lru_prep(
    const float* __restrict__ B_re, const float* __restrict__ B_im,
    const float* __restrict__ C_im,
    const float* __restrict__ nu_log, const float* __restrict__ theta_log,
    float* __restrict__ Bn_re, float* __restrict__ Bn_im,
    float* __restrict__ C_imn,
    float* __restrict__ Lam_re, float* __restrict__ Lam_im) {
  int t = blockIdx.x * 256 + threadIdx.x;   // 0 .. 65535
  int n = t >> 8;
  int h = t & 255;

  float nu    = __expf(nu_log[n]);
  float th    = __expf(theta_log[n]);
  float mag   = __expf(-nu);                 // |Lambda|
  float gamma = sqrtf(fmaxf(1.0f - mag * mag, 0.0f));

  Bn_re[t] = B_re[t] * gamma;                // B_norm[n][h], native layout
  Bn_im[t] = B_im[t] * gamma;
  C_imn[t] = -C_im[t];                       // -C_im[h][n], native layout

  if (h == 0) {
    Lam_re[n] = mag * __cosf(th);
    Lam_im[n] = mag * __sinf(th);
  }
}

// ---------------------------------------------------------------------------
// K2: Bu = x @ Bn^T (fp32 WMMA). Block owns a 16-row m-tile. x strip +
// double-buffered 32-deep B panels (re+im, n-major) are TDM-loaded into LDS;
// wave 0 drives the DMA. Inner loop: 5x ds_load_b64 + 4x wmma per K-step.
// ---------------------------------------------------------------------------
__global__ __launch_bounds__(256) void lru_gemm_bu(
    const float* __restrict__ x,
    const float* __restrict__ Bn_re, const float* __restrict__ Bn_im,
    float* __restrict__ Bu_re, float* __restrict__ Bu_im) {
  __shared__ __align__(16) float As[16 * PITCH];        // 16.6 KB
  __shared__ __align__(16) float Bs[2][2][NDIM * KP];   // [buf][plane], 139 KB

  int mt  = blockIdx.x;                       // 0 .. 2047
  int tid = threadIdx.x;
  bool issuer = (tid < 32);                   // wave 0 drives TDM

  if (issuer) {
    tdm_load_2d(lds_off(As), x + (size_t)mt * 16 * HDIM,
                HDIM, 16, HDIM, /*icode=*/7, /*acode=*/3);   // pitch 260
    tdm_load_panel(lds_off(Bs[0][0]), Bn_re);
    tdm_load_panel(lds_off(Bs[0][1]), Bn_im);
  }

  int wave = tid >> 5;
  int lane = tid & 31;
  int hl   = lane >> 4;                       // lane half (K split, f32 frags)
  int r    = lane & 15;
  int n0   = wave * 32;                       // this wave's first n-tile

  v8f cre0 = {}, cim0 = {}, cre1 = {}, cim1 = {};

  for (int kc = 0; kc < HDIM / KC; ++kc) {
    if (issuer) __builtin_amdgcn_s_wait_tensorcnt(0);
    __syncthreads();                          // panel kc visible to all waves
    if (issuer && (kc + 1) < HDIM / KC) {     // prefetch next panel
      int nb = (kc + 1) & 1;
      tdm_load_panel(lds_off(Bs[nb][0]), Bn_re + (size_t)(kc + 1) * KC);
      tdm_load_panel(lds_off(Bs[nb][1]), Bn_im + (size_t)(kc + 1) * KC);
    }
    const float* bre = Bs[kc & 1][0];
    const float* bim = Bs[kc & 1][1];

#pragma unroll 4
    for (int kk = 0; kk < KC; kk += 4) {
      int ks = kk + 2 * hl;                   // this half's K pair
      v2f a   = ldv2(&As[r * PITCH + kc * KC + ks]);          // A(m=r, k..k+1)
      v2f br0 = ldv2(&bre[(n0 + r) * KP + ks]);               // B(k..k+1, n)
      v2f bi0 = ldv2(&bim[(n0 + r) * KP + ks]);
      v2f br1 = ldv2(&bre[(n0 + 16 + r) * KP + ks]);
      v2f bi1 = ldv2(&bim[(n0 + 16 + r) * KP + ks]);

      cre0 = wmma_f32(a, br0, cre0);
      cim0 = wmma_f32(a, bi0, cim0);
      cre1 = wmma_f32(a, br1, cre1);
      cim1 = wmma_f32(a, bi1, cim1);
    }
  }

  // C/D layout: VGPR i -> row (i + 8*hl), lane r -> col.
#pragma unroll
  for (int i = 0; i < 8; ++i) {
    size_t row = (size_t)mt * 16 + i + hl * 8;
    size_t i0  = row * NDIM + n0 + r;
    Bu_re[i0]      = cre0[i];
    Bu_im[i0]      = cim0[i];
    Bu_re[i0 + 16] = cre1[i];
    Bu_im[i0 + 16] = cim1[i];
  }
}

// ---------------------------------------------------------------------------
// K3: per-(b, chunk, n) local scan from zero -> chunk final state F.
// ---------------------------------------------------------------------------
__global__ __launch_bounds__(256) void lru_chunk_reduce(
    const float* __restrict__ Bu_re, const float* __restrict__ Bu_im,
    const float* __restrict__ Lam_re, const float* __restrict__ Lam_im,
    float* __restrict__ F_re, float* __restrict__ F_im) {
  int t = blockIdx.x * 256 + threadIdx.x;  // 0 .. 65535
  int n = t & 255;
  int c = (t >> 8) & (NCHUNK - 1);
  int b = t >> 13;

  float lre = Lam_re[n], lim = Lam_im[n];
  float sre = 0.0f, sim = 0.0f;
  size_t base = ((size_t)b * SEQL + (size_t)c * CHUNK) * NDIM + n;
  for (int l = 0; l < CHUNK; ++l) {
    float bre = Bu_re[base + (size_t)l * NDIM];
    float bim = Bu_im[base + (size_t)l * NDIM];
    float nre = fmaf(lre, sre, fmaf(-lim, sim, bre));
    float nim = fmaf(lre, sim, fmaf( lim, sre, bim));
    sre = nre; sim = nim;
  }
  F_re[t] = sre;
  F_im[t] = sim;
}

// ---------------------------------------------------------------------------
// K4: sequential scan over the 32 chunk carries per (b, n).
// ---------------------------------------------------------------------------
__global__ __launch_bounds__(256) void lru_carry_scan(
    const float* __restrict__ F_re, const float* __restrict__ F_im,
    const float* __restrict__ nu_log, const float* __restrict__ theta_log,
    float* __restrict__ P_re, float* __restrict__ P_im) {
  int t = blockIdx.x * 256 + threadIdx.x;  // 0 .. 2047
  int n = t & 255;
  int b = t >> 8;

  float nu  = __expf(nu_log[n]);
  float th  = __expf(theta_log[n]);
  float mag = __expf(-(float)CHUNK * nu);       // |Lambda|^128
  float are = mag * __cosf((float)CHUNK * th);
  float aim = mag * __sinf((float)CHUNK * th);

  float cre = 0.0f, cim = 0.0f;
  for (int c = 0; c < NCHUNK; ++c) {
    int idx = (b * NCHUNK + c) * NDIM + n;
    P_re[idx] = cre;
    P_im[idx] = cim;
    float fre = F_re[idx], fim = F_im[idx];
    float nre = fmaf(are, cre, fmaf(-aim, cim, fre));
    float nim = fmaf(are, cim, fmaf( aim, cre, fim));
    cre = nre; cim = nim;
  }
}

// ---------------------------------------------------------------------------
// K5: fused final scan + GEMM2 + skip. Block owns (b, chunk): thread n
// rescans 16 rows at a time into padded LDS tiles (re/im); 8 waves consume
// them as WMMA A-fragments vs TDM-staged, double-buffered C panels (native
// [h][n] order -> K-contiguous b64 fragments). x*D skip at store; states
// never touch HBM.
// ---------------------------------------------------------------------------
__global__ __launch_bounds__(256) void lru_scan_gemm_y(
    const float* __restrict__ Bu_re, const float* __restrict__ Bu_im,
    const float* __restrict__ P_re, const float* __restrict__ P_im,
    const float* __restrict__ Lam_re, const float* __restrict__ Lam_im,
    const float* __restrict__ C_re, const float* __restrict__ C_imn,
    const float* __restrict__ x, const float* __restrict__ D,
    float* __restrict__ y) {
  __shared__ __align__(16) float Sre[16 * PITCH];
  __shared__ __align__(16) float Sim[16 * PITCH];
  __shared__ __align__(16) float Cs[2][2][HDIM * KP];   // [buf][plane], 139 KB

  int b   = blockIdx.x >> 5;                  // / NCHUNK
  int c   = blockIdx.x & (NCHUNK - 1);
  int tid = threadIdx.x;
  int n   = tid;                              // scan lane: one state dim/thread
  bool issuer = (tid < 32);

  float sre = P_re[(b * NCHUNK + c) * NDIM + n];
  float sim = P_im[(b * NCHUNK + c) * NDIM + n];
  float lre = Lam_re[n], lim = Lam_im[n];

  int wave = tid >> 5;
  int lane = tid & 31;
  int hl   = lane >> 4;
  int r    = lane & 15;
  int h0   = wave * 32;                       // this wave's first h-tile

  size_t rowbase = (size_t)b * SEQL + (size_t)c * CHUNK;

  for (int sb = 0; sb < CHUNK / 16; ++sb) {
    // kick off first C panel; DMA overlaps the serial scan below
    if (issuer) {
      tdm_load_panel(lds_off(Cs[0][0]), C_re);
      tdm_load_panel(lds_off(Cs[0][1]), C_imn);
    }

    // --- scan 16 steps, depositing states into LDS ---
    for (int j = 0; j < 16; ++j) {
      size_t idx = (rowbase + sb * 16 + j) * NDIM + n;
      float bre = Bu_re[idx];
      float bim = Bu_im[idx];
      float nre = fmaf(lre, sre, fmaf(-lim, sim, bre));
      float nim = fmaf(lre, sim, fmaf( lim, sre, bim));
      sre = nre; sim = nim;
      Sre[j * PITCH + n] = sre;
      Sim[j * PITCH + n] = sim;
    }

    // --- GEMM: y[16 x 256] tile, K = n-dim, both planes into one acc ---
    v8f acc0 = {}, acc1 = {};
    for (int kc = 0; kc < NDIM / KC; ++kc) {
      if (issuer) __builtin_amdgcn_s_wait_tensorcnt(0);
      __syncthreads();                        // C panel kc + scan tile ready
      if (issuer && (kc + 1) < NDIM / KC) {
        int nb = (kc + 1) & 1;
        tdm_load_panel(lds_off(Cs[nb][0]), C_re  + (size_t)(kc + 1) * KC);
        tdm_load_panel(lds_off(Cs[nb][1]), C_imn + (size_t)(kc + 1) * KC);
      }
      const float* bre = Cs[kc & 1][0];
      const float* bim = Cs[kc & 1][1];

#pragma unroll 4
      for (int kk = 0; kk < KC; kk += 4) {
        int ks = kk + 2 * hl;                 // this half's K pair
        v2f ar  = ldv2(&Sre[r * PITCH + kc * KC + ks]);
        v2f ai  = ldv2(&Sim[r * PITCH + kc * KC + ks]);
        v2f br0 = ldv2(&bre[(h0 + r) * KP + ks]);
        v2f bi0 = ldv2(&bim[(h0 + r) * KP + ks]);
        v2f br1 = ldv2(&bre[(h0 + 16 + r) * KP + ks]);
        v2f bi1 = ldv2(&bim[(h0 + 16 + r) * KP + ks]);

        acc0 = wmma_f32(ar, br0, acc0);
        acc0 = wmma_f32(ai, bi0, acc0);       // C_im already negated
        acc1 = wmma_f32(ar, br1, acc1);
        acc1 = wmma_f32(ai, bi1, acc1);
      }
    }

    // --- store with x*D skip ---
#pragma unroll
    for (int i = 0; i < 8; ++i) {
      size_t row = rowbase + sb * 16 + i + hl * 8;
      int col0 = h0 + r;
      int col1 = col0 + 16;
      y[row * HDIM + col0] = acc0[i] + x[row * HDIM + col0] * D[col0];
      y[row * HDIM + col1] = acc1[i] + x[row * HDIM + col1] * D[col1];
    }
    __syncthreads();   // all waves done reading Sre/Sim before next scan
  }
}

// ---------------------------------------------------------------------------
// Host launcher
// ---------------------------------------------------------------------------
extern "C" void kernel_launch(void* const* d_in, const int* in_sizes, int n_in,
                              void* d_out, int out_size, void* d_ws, size_t ws_size,
                              hipStream_t stream) {
  const float* x         = (const float*)d_in[0];
  const float* B_re      = (const float*)d_in[1];
  const float* B_im      = (const float*)d_in[2];
  const float* C_re      = (const float*)d_in[3];
  const float* C_im      = (const float*)d_in[4];
  const float* nu_log    = (const float*)d_in[5];
  const float* theta_log = (const float*)d_in[6];
  const float* Dvec      = (const float*)d_in[7];
  float* y = (float*)d_out;

  // workspace carve-up (floats); total ~17.1M floats (~68.6 MB)
  float* ws     = (float*)d_ws;
  float* Bn_re  = ws;
  float* Bn_im  = Bn_re + NDIM * HDIM;
  float* C_imn  = Bn_im + NDIM * HDIM;
  float* Lam_re = C_imn + NDIM * HDIM;
  float* Lam_im = Lam_re + NDIM;
  float* Bu_re  = Lam_im + NDIM;
  float* Bu_im  = Bu_re  + (size_t)MTOT * NDIM;
  float* F_re   = Bu_im  + (size_t)MTOT * NDIM;
  float* F_im   = F_re   + BSZ * NCHUNK * NDIM;
  float* P_re   = F_im   + BSZ * NCHUNK * NDIM;
  float* P_im   = P_re   + BSZ * NCHUNK * NDIM;

  lru_prep<<<(NDIM * HDIM) / 256, 256, 0, stream>>>(
      B_re, B_im, C_im, nu_log, theta_log,
      Bn_re, Bn_im, C_imn, Lam_re, Lam_im);

  lru_gemm_bu<<<MTOT / 16, 256, 0, stream>>>(x, Bn_re, Bn_im, Bu_re, Bu_im);

  lru_chunk_reduce<<<(BSZ * NCHUNK * NDIM) / 256, 256, 0, stream>>>(
      Bu_re, Bu_im, Lam_re, Lam_im, F_re, F_im);

  lru_carry_scan<<<(BSZ * NDIM) / 256, 256, 0, stream>>>(
      F_re, F_im, nu_log, theta_log, P_re, P_im);

  lru_scan_gemm_y<<<BSZ * NCHUNK, 256, 0, stream>>>(
      Bu_re, Bu_im, P_re, P_im, Lam_re, Lam_im, C_re, C_imn, x, Dvec, y);
}